// ZeroVelocityWrapper_38482906972738
// MI455X (gfx1250) — compile-verified
//
#include <hip/hip_runtime.h>
#include <hip/hip_bf16.h>

// Problem constants (from reference setup_inputs)
#define B_SZ   16
#define T_IN   64
#define N_J    52
#define T_OUT  4096
#define T_PAD  (T_OUT + T_IN)          // 4160 = 63 front pads + 4096 core + 1 back pad
#define ROT_R  (B_SZ * N_J * 6)        // 4992 rows of the rotation GEMM
#define ROOT_R (B_SZ * 3)              // 48 rows of the root GEMM
#define ROT_TSTRIDE (N_J * 6)          // 312 floats between consecutive t for fixed (b,n,c)

typedef __attribute__((ext_vector_type(2))) float v2f;
typedef __attribute__((ext_vector_type(8))) float v8f;

// ---------------------------------------------------------------------------
// Kernel 1: exact softmax weights.  With TEMP=1e5, exp(TEMP*(d - dmax)) is
// exactly 1.0f when d==dmax and exactly 0.0f otherwise (fp32 underflow), so
// weights[t,s] = 1/k over the k duplicates of the nearest past keyframe.
// Layout: W[t * T_OUT + s]  (row-major [T_in, T_out]) -> lane-contiguous in s.
// ---------------------------------------------------------------------------
__global__ void zv_weights_kernel(const int* __restrict__ in_idx,
                                  const int* __restrict__ tgt_idx,
                                  float* __restrict__ W) {
    int s = blockIdx.x * blockDim.x + threadIdx.x;
    if (s >= T_OUT) return;
    int tgt = tgt_idx[s];
    int sel = -1, k = 0;
#pragma unroll
    for (int t = 0; t < T_IN; ++t) {
        int v = in_idx[t];
        if (v <= tgt) {
            if (v > sel) { sel = v; k = 1; }
            else if (v == sel) { ++k; }
        }
    }
    float w = 1.0f / (float)k;  // k >= 1 guaranteed (idx[0] == 0)
#pragma unroll
    for (int t = 0; t < T_IN; ++t) {
        int v = in_idx[t];
        W[t * T_OUT + s] = (v == sel) ? w : 0.0f;
    }
}

// ---------------------------------------------------------------------------
// Kernel 2: rotation einsum 'btnc,ts->bsnc' as WMMA f32 GEMM, transposed form:
//   D[s, r] = sum_t W[t, s] * A[r, t],  r = (b, n, c) flattened (4992 rows).
// One wave32 per 16x16 tile of D.  Lane -> r (contiguous loads AND stores),
// 16x WMMA_F32_16X16X4_F32 covers K=64.  Core frames written directly into
// the padded output at time offset +63.
// ---------------------------------------------------------------------------
__global__ void zv_rot_gemm_kernel(const float* __restrict__ A,
                                   const float* __restrict__ W,
                                   float* __restrict__ jr_out) {
    const int lane  = threadIdx.x & 31;
    const int mn    = lane & 15;        // A'-operand row (s offset) / B'-operand col (r offset)
    const int khalf = lane >> 4;        // 0: K pair {0,1}; 1: K pair {2,3} within each K=4 step
    const int s0 = blockIdx.x * 16;
    const int r0 = blockIdx.y * 16;

    // This lane's column r = (b, n, c)
    const int rb  = r0 + mn;
    const int b   = rb / ROT_TSTRIDE;
    const int rem = rb % ROT_TSTRIDE;
    const int jn  = rem / 6;
    const int c   = rem % 6;
    const float* __restrict__ Ab = A + ((size_t)b * T_IN * N_J + jn) * 6 + c; // t stride = 312
    const float* __restrict__ Wb = W + s0 + mn;                               // t stride = T_OUT

    v8f acc = {0.f, 0.f, 0.f, 0.f, 0.f, 0.f, 0.f, 0.f};
#pragma unroll
    for (int kk = 0; kk < T_IN; kk += 4) {
        const int ka = kk + 2 * khalf;
        v2f av, bv;
        av.x = Wb[(size_t)ka * T_OUT];          // A' = W^T : [16 s-rows x 4 k]
        av.y = Wb[(size_t)(ka + 1) * T_OUT];
        bv.x = Ab[(size_t)ka * ROT_TSTRIDE];    // B' = A^T : [4 k x 16 r-cols]
        bv.y = Ab[(size_t)(ka + 1) * ROT_TSTRIDE];
        acc = __builtin_amdgcn_wmma_f32_16x16x4_f32(
            /*neg_a=*/false, av, /*neg_b=*/false, bv,
            /*c_mod=*/(short)0, acc, /*reuse_a=*/false, /*reuse_b=*/false);
    }

    // D rows: VGPR i -> s = s0 + i (+8 for lanes 16-31); column fixed = this lane's r.
#pragma unroll
    for (int i = 0; i < 8; ++i) {
        const size_t u = (size_t)(s0 + 8 * khalf + i) + (T_IN - 1); // +63 pad offset
        jr_out[(((size_t)b * T_PAD + u) * N_J + jn) * 6 + c] = acc[i];
    }
}

// ---------------------------------------------------------------------------
// Kernel 3: root einsum 'btc,ts->bsc' (48 x 64 x 4096) + broadcast over 52
// joints, written directly into the padded jp region at time offset +63.
// ---------------------------------------------------------------------------
__global__ void zv_root_gemm_kernel(const float* __restrict__ A,
                                    const float* __restrict__ W,
                                    float* __restrict__ jp_out) {
    const int lane  = threadIdx.x & 31;
    const int mn    = lane & 15;
    const int khalf = lane >> 4;
    const int s0 = blockIdx.x * 16;
    const int r0 = blockIdx.y * 16;

    const int rb = r0 + mn;         // r = (b, c), 48 rows
    const int b  = rb / 3;
    const int c  = rb % 3;
    const float* __restrict__ Ab = A + (size_t)b * T_IN * 3 + c;  // t stride = 3
    const float* __restrict__ Wb = W + s0 + mn;

    v8f acc = {0.f, 0.f, 0.f, 0.f, 0.f, 0.f, 0.f, 0.f};
#pragma unroll
    for (int kk = 0; kk < T_IN; kk += 4) {
        const int ka = kk + 2 * khalf;
        v2f av, bv;
        av.x = Wb[(size_t)ka * T_OUT];
        av.y = Wb[(size_t)(ka + 1) * T_OUT];
        bv.x = Ab[(size_t)ka * 3];
        bv.y = Ab[(size_t)(ka + 1) * 3];
        acc = __builtin_amdgcn_wmma_f32_16x16x4_f32(
            false, av, false, bv, (short)0, acc, false, false);
    }

#pragma unroll
    for (int i = 0; i < 8; ++i) {
        const size_t u = (size_t)(s0 + 8 * khalf + i) + (T_IN - 1);
        const size_t base = ((size_t)b * T_PAD + u) * N_J;
        for (int n2 = 0; n2 < N_J; ++n2)          // broadcast root over joints
            jp_out[(base + n2) * 3 + c] = acc[i];
    }
}

// ---------------------------------------------------------------------------
// Kernel 4: fill pad frames: u in [0,63) <- core frame s=0 (u=63),
//           u = T_PAD-1 <- core frame s=T_OUT-1 (u=T_PAD-2).
// Runs after the GEMMs on the same in-order stream.
// ---------------------------------------------------------------------------
#define JP_PAD (B_SZ * T_IN * N_J * 3)   // 159744
#define JR_PAD (B_SZ * T_IN * N_J * 6)   // 319488

__global__ void zv_pad_kernel(float* __restrict__ jp, float* __restrict__ jr) {
    int idx = blockIdx.x * blockDim.x + threadIdx.x;
    if (idx < JP_PAD) {
        int c  = idx % 3;   int t1 = idx / 3;
        int n  = t1 % N_J;  int t2 = t1 / N_J;
        int j  = t2 % T_IN; int b  = t2 / T_IN;
        int ud = (j < T_IN - 1) ? j : (T_PAD - 1);
        int us = (j < T_IN - 1) ? (T_IN - 1) : (T_PAD - 2);
        jp[(((size_t)b * T_PAD + ud) * N_J + n) * 3 + c] =
            jp[(((size_t)b * T_PAD + us) * N_J + n) * 3 + c];
    } else if (idx < JP_PAD + JR_PAD) {
        int q  = idx - JP_PAD;
        int c  = q % 6;     int t1 = q / 6;
        int n  = t1 % N_J;  int t2 = t1 / N_J;
        int j  = t2 % T_IN; int b  = t2 / T_IN;
        int ud = (j < T_IN - 1) ? j : (T_PAD - 1);
        int us = (j < T_IN - 1) ? (T_IN - 1) : (T_PAD - 2);
        jr[(((size_t)b * T_PAD + ud) * N_J + n) * 6 + c] =
            jr[(((size_t)b * T_PAD + us) * N_J + n) * 6 + c];
    }
}

extern "C" void kernel_launch(void* const* d_in, const int* in_sizes, int n_in,
                              void* d_out, int out_size, void* d_ws, size_t ws_size,
                              hipStream_t stream) {
    const float* root   = (const float*)d_in[0];   // [16, 64, 3]
    const float* rot    = (const float*)d_in[1];   // [16, 64, 52, 6]
    const int*   in_idx = (const int*)d_in[2];     // [64]
    const int*   tg_idx = (const int*)d_in[3];     // [4096]

    float* jp = (float*)d_out;                                    // [16,4160,52,3]
    float* jr = jp + (size_t)B_SZ * T_PAD * N_J * 3;              // [16,4160,52,6]
    float* W  = (float*)d_ws;                                     // [64,4096] = 1 MB

    // 1) exact weights
    zv_weights_kernel<<<(T_OUT + 255) / 256, 256, 0, stream>>>(in_idx, tg_idx, W);

    // 2) rotation GEMM: 256 s-tiles x 312 r-tiles, one wave32 per 16x16 tile
    zv_rot_gemm_kernel<<<dim3(T_OUT / 16, ROT_R / 16), 32, 0, stream>>>(rot, W, jr);

    // 3) root GEMM + joint broadcast: 256 s-tiles x 3 r-tiles
    zv_root_gemm_kernel<<<dim3(T_OUT / 16, ROOT_R / 16), 32, 0, stream>>>(root, W, jp);

    // 4) fill pad frames (reads core frames written above; stream is in-order)
    zv_pad_kernel<<<(JP_PAD + JR_PAD + 255) / 256, 256, 0, stream>>>(jp, jr);
}